// LightGCN_53626961658032
// MI455X (gfx1250) — compile-verified
//
#include <hip/hip_runtime.h>
#include <hip/hip_bf16.h>
#include <stdint.h>

// LightGCN propagation on MI455X (gfx1250).
//
// Roofline: ~0.3 GFLOP vs ~1.4 GB of gather/scatter traffic -> ~0.2 FLOP/B.
// WMMA is irrelevant; the working set (25.6MB features + 6.4MB edges) lives in
// the 192MB L2, so the kernel is L2-gather/atomic bound. We use the CDNA5
// Tensor Data Mover in *gather mode* (hardware row-gather into LDS, 8 rows per
// 32-bit-index descriptor) with TENSORcnt software pipelining, and native
// global_atomic_add_f32 for the scatter.

#define DIMF      64
#define ALPHA     0.25f
#define NLAYERS   3
#define EPB       32   // edges per workgroup (one wave)
#define BATCH     8    // edges per TDM gather descriptor (32-bit index max)

typedef unsigned int v4u __attribute__((ext_vector_type(4)));
typedef int          v4i __attribute__((ext_vector_type(4)));
typedef int          v8i __attribute__((ext_vector_type(8)));

__device__ __forceinline__ void atomic_add_f32(float* p, float v) {
  // Hardware global_atomic_add_f32 (non-returning), agent scope.
  unsafeAtomicAdd(p, v);
}

// s_wait_tensorcnt requires an integer constant expression argument; a
// non-type template parameter satisfies that even inside unrolled code.
template <short CNT>
__device__ __forceinline__ void wait_tensorcnt() {
  __builtin_amdgcn_s_wait_tensorcnt(CNT);
}

// Issue one TDM gather: fetch `nvalid` rows (64 fp32 each) of the row-major
// tensor `base` (n_rows x 64) into LDS at byte offset lds_byte_off.
// Row indices (32-bit mode) in idx_lo/idx_hi. D# packing per ISA 08 §8.
__device__ __forceinline__ void tdm_gather_rows(const float* __restrict__ base,
                                                unsigned lds_byte_off,
                                                v4i idx_lo, v4i idx_hi,
                                                int nvalid, int n_rows) {
  unsigned long long ga = (unsigned long long)(uintptr_t)base;
  v4u g0;
  g0.x = 0xC0000001u;                                        // count=1 | gather_index_size=32b | gather_mode=1
  g0.y = lds_byte_off;                                       // lds_addr [63:32]
  g0.z = (unsigned)(ga & 0xFFFFFFFFu);                       // global_addr [95:64]
  g0.w = (unsigned)((ga >> 32) & 0x1FFFFFFu) | 0x80000000u;  // global_addr hi | type=2
  v8i g1;
  g1[0] = (int)(2u << 16);                                   // workgroup_mask=0 | data_size=4B
  g1[1] = (int)((unsigned)DIMF << 16);                       // tensor_dim0 lo16 = 64
  g1[2] = (int)(((unsigned)n_rows & 0xFFFFu) << 16);         // dim0 hi16=0 | tensor_dim1 lo16
  g1[3] = (int)(((unsigned)DIMF << 16) |
                ((unsigned)n_rows >> 16));                   // tensor_dim1 hi16 | tile_dim0=64
  g1[4] = (int)((unsigned)nvalid & 0xFFFFu);                 // tile_dim1 = #valid indices
  g1[5] = DIMF;                                              // tensor_dim0_stride = 64 elements
  g1[6] = 0;
  g1[7] = 0;
#if __clang_major__ >= 23
  v8i gx = {0, 0, 0, 0, 0, 0, 0, 0};
  __builtin_amdgcn_tensor_load_to_lds(g0, g1, idx_lo, idx_hi, gx, 0);
#else
  __builtin_amdgcn_tensor_load_to_lds(g0, g1, idx_lo, idx_hi, 0);
#endif
}

// ---------------- elementwise helpers ----------------

__global__ void zero4_kernel(float4* __restrict__ p, int n4) {
  int i = blockIdx.x * blockDim.x + threadIdx.x;
  if (i < n4) p[i] = make_float4(0.f, 0.f, 0.f, 0.f);
}

__global__ void deg_kernel(const int* __restrict__ col, float* __restrict__ deg,
                           int E) {
  int i = blockIdx.x * blockDim.x + threadIdx.x;
  if (i < E) atomic_add_f32(&deg[col[i]], 1.0f);
}

__global__ void dinv_kernel(const float* __restrict__ deg,
                            float* __restrict__ dinv, int N) {
  int i = blockIdx.x * blockDim.x + threadIdx.x;
  if (i < N) {
    float d = deg[i];
    dinv[i] = (d > 0.f) ? __frsqrt_rn(d) : 0.f;
  }
}

__global__ void init_out_kernel(float4* __restrict__ out,
                                const float4* __restrict__ x, int n4) {
  int i = blockIdx.x * blockDim.x + threadIdx.x;
  if (i < n4) {
    float4 v = x[i];
    out[i] = make_float4(ALPHA * v.x, ALPHA * v.y, ALPHA * v.z, ALPHA * v.w);
  }
}

__global__ void axpy_out_kernel(float4* __restrict__ out,
                                const float4* __restrict__ xl, int n4) {
  int i = blockIdx.x * blockDim.x + threadIdx.x;
  if (i < n4) {
    float4 o = out[i];
    float4 v = xl[i];
    o.x += ALPHA * v.x; o.y += ALPHA * v.y;
    o.z += ALPHA * v.z; o.w += ALPHA * v.w;
    out[i] = o;
  }
}

// ---------------- SpMM layer: x_next[col] += dinv[row]*dinv[col]*x[row] ----
//
// One wave (32 lanes) per 32 edges. Phase 1: issue 4 TDM gather descriptors
// (8 source rows each) into LDS. Phase 2: drain in order with progressive
// s_wait_tensorcnt, scale each gathered row by its edge weight, and scatter
// with global_atomic_add_f32 (lane handles 2 of the 64 dims).

__global__ void __launch_bounds__(32)
spmm_tdm_kernel(const int* __restrict__ row, const int* __restrict__ col,
                const float* __restrict__ dinv, const float* __restrict__ xin,
                float* __restrict__ xout, int E, int N) {
  __shared__ float smem[EPB * DIMF];  // 8 KB: 32 gathered rows
  const int base0 = blockIdx.x * EPB;
  // Low 32 bits of the generic (flat-aperture) shared pointer are the
  // workgroup-relative LDS byte offset (ISA 10.2 aperture encoding).
  const unsigned lds0 = (unsigned)(unsigned long long)(uintptr_t)&smem[0];
  const int lane = threadIdx.x;

  // ---- issue 4 gather descriptors (TENSORcnt -> 4) ----
#pragma unroll
  for (int b = 0; b < 4; ++b) {
    const int base = base0 + b * BATCH;
    const int nvalid = min(BATCH, E - base);
    if (nvalid > 0) {
      v4i lo, hi;
#pragma unroll
      for (int k = 0; k < 4; ++k) {
        lo[k] = row[min(base + k,     E - 1)];
        hi[k] = row[min(base + 4 + k, E - 1)];
      }
      tdm_gather_rows(xin, lds0 + (unsigned)(b * BATCH * DIMF * 4),
                      lo, hi, nvalid, N);
    }
  }

  // Per-batch drain: scale gathered rows by edge weight and scatter-add.
  auto drain = [&](int b) {
    const int base = base0 + b * BATCH;
#pragma unroll
    for (int e = 0; e < BATCH; ++e) {
      const int idx = base + e;
      if (idx < E) {
        const int r = row[idx];             // uniform -> scalar loads
        const int c = col[idx];
        const float w = dinv[r] * dinv[c];  // edge weight
        const float2 v =
            *(const float2*)&smem[(b * BATCH + e) * DIMF + lane * 2];
        float* dst = xout + (size_t)c * DIMF + (size_t)(lane * 2);
        atomic_add_f32(dst + 0, w * v.x);
        atomic_add_f32(dst + 1, w * v.y);
      }
    }
  };

  // ---- drain pipeline: batch b is done once TENSORcnt <= 3-b (in-order) ----
  wait_tensorcnt<3>(); drain(0);
  wait_tensorcnt<2>(); drain(1);
  wait_tensorcnt<1>(); drain(2);
  wait_tensorcnt<0>(); drain(3);
}

// ---------------- host-side orchestration ----------------

extern "C" void kernel_launch(void* const* d_in, const int* in_sizes, int n_in,
                              void* d_out, int out_size, void* d_ws,
                              size_t ws_size, hipStream_t stream) {
  const float* x = (const float*)d_in[0];        // [N, 64] fp32
  const int* ei = (const int*)d_in[1];           // [2, E] int
  const int N = in_sizes[0] / DIMF;              // 100000
  const int E = in_sizes[1] / 2;                 // 800000
  const int* row = ei;                           // edge_index[0] = src
  const int* col = ei + E;                       // edge_index[1] = dst
  float* out = (float*)d_out;

  float* ws   = (float*)d_ws;
  float* deg  = ws;                              // [N]
  float* dinv = ws + N;                          // [N]
  float* xA   = ws + 2 * (size_t)N;              // [N,64]
  float* xB   = xA + (size_t)N * DIMF;           // [N,64]

  const int nf  = N * DIMF;                      // 6.4M floats
  const int nf4 = nf / 4;                        // 1.6M float4
  const int THREADS = 256;
  const int gN4  = (N / 4 + THREADS - 1) / THREADS;
  const int gN   = (N + THREADS - 1) / THREADS;
  const int gE   = (E + THREADS - 1) / THREADS;
  const int gF4  = (nf4 + THREADS - 1) / THREADS;
  const int gSp  = (E + EPB - 1) / EPB;          // 25000 waves

  // Normalization: deg (over dst), then dinv = rsqrt(deg).
  zero4_kernel<<<gN4, THREADS, 0, stream>>>((float4*)deg, N / 4);
  deg_kernel<<<gE, THREADS, 0, stream>>>(col, deg, E);
  dinv_kernel<<<gN, THREADS, 0, stream>>>(deg, dinv, N);

  // out = alpha * x
  init_out_kernel<<<gF4, THREADS, 0, stream>>>((float4*)out, (const float4*)x,
                                               nf4);

  // 3 propagation layers, ping-ponging xA/xB; layer 0 reads the input x.
  const float* cur = x;
  float* bufs[2] = {xA, xB};
  for (int l = 0; l < NLAYERS; ++l) {
    float* nxt = bufs[l & 1];
    zero4_kernel<<<gF4, THREADS, 0, stream>>>((float4*)nxt, nf4);
    spmm_tdm_kernel<<<gSp, EPB, 0, stream>>>(row, col, dinv, cur, nxt, E, N);
    axpy_out_kernel<<<gF4, THREADS, 0, stream>>>((float4*)out,
                                                 (const float4*)nxt, nf4);
    cur = nxt;
  }
}